// SpatioGATLayer_88184268521806
// MI455X (gfx1250) — compile-verified
//
#include <hip/hip_runtime.h>
#include <math.h>

#define BB 4
#define NN 50000
#define EE 800000
#define EF 10
#define EH 4
#define CC 64
#define ET (EE + NN)          // edges + self loops
#define TPB (NN / 16)         // 3125 row-tiles per batch (N divisible by 16)

typedef __attribute__((ext_vector_type(2))) float v2f;
typedef __attribute__((ext_vector_type(8))) float v8f;

// ---- order-preserving float <-> uint transform for atomicMax on float ----
__device__ __forceinline__ unsigned f2o(float f) {
    unsigned u = __float_as_uint(f);
    return (u & 0x80000000u) ? ~u : (u | 0x80000000u);
}
__device__ __forceinline__ float o2f(unsigned u) {
    return (u & 0x80000000u) ? __uint_as_float(u & 0x7FFFFFFFu)
                             : __uint_as_float(~u);
}

// ---- v_edge[j] = sum_c lin_edge_W[j,c] * att_edge[c]  (collapse of two matmuls) ----
__global__ void k_vedge(const float* __restrict__ linW, const float* __restrict__ att_e,
                        float* __restrict__ v_edge) {
    int j = threadIdx.x;
    if (j < EH) {
        float s = 0.f;
        for (int c = 0; c < CC; ++c) s += linW[j * CC + c] * att_e[c];
        v_edge[j] = s;
    }
}

// ---- edge MLP + per-dst accumulation for self-loop mean + edge logit term ----
__global__ void k_edge_mlp(const float* __restrict__ ea, const int* __restrict__ eidx,
                           const float* __restrict__ W1, const float* __restrict__ W2,
                           const float* __restrict__ v_edge,
                           float* __restrict__ e_out, float* __restrict__ cnt,
                           float* __restrict__ lacc, float* __restrict__ a_edge) {
    int i = blockIdx.x * blockDim.x + threadIdx.x;
    if (i >= EE) return;
    float t[EH];
    for (int j = 0; j < EH; ++j) t[j] = 0.f;
    for (int k = 0; k < EF; ++k) {
        float a = ea[(size_t)i * EF + k];
        for (int j = 0; j < EH; ++j) t[j] += a * W1[k * EH + j];
    }
    for (int j = 0; j < EH; ++j) t[j] = (t[j] > 0.f) ? t[j] : 0.01f * t[j];
    float ev[EH];
    float dotv = 0.f;
    for (int j2 = 0; j2 < EH; ++j2) {
        float s = 0.f;
        for (int j = 0; j < EH; ++j) s += t[j] * W2[j * EH + j2];
        ev[j2] = s;
        e_out[(size_t)i * EH + j2] = s;
        dotv += s * v_edge[j2];
    }
    a_edge[i] = dotv;
    int dst = eidx[EE + i];  // edge_index row 1
    atomicAdd(&cnt[dst], 1.f);
    for (int j = 0; j < EH; ++j) atomicAdd(&lacc[dst * EH + j], ev[j]);
}

// ---- self-loop attr mean -> its logit contribution ----
__global__ void k_loop(const float* __restrict__ lacc, const float* __restrict__ cnt,
                       const float* __restrict__ v_edge, float* __restrict__ a_edge) {
    int n = blockIdx.x * blockDim.x + threadIdx.x;
    if (n >= NN) return;
    float c = fmaxf(cnt[n], 1.f);
    float s = 0.f;
    for (int j = 0; j < EH; ++j) s += (lacc[n * EH + j] / c) * v_edge[j];
    a_edge[EE + n] = s;
}

// ---- h = x @ W via V_WMMA_F32_16X16X4_F32 (one wave -> 16x64 tile) ----
__global__ void __launch_bounds__(256) k_h_gemm(const float* __restrict__ x,
                                                const float* __restrict__ W,
                                                float* __restrict__ h) {
    __shared__ float sW[CC * CC];
    for (int i = threadIdx.x; i < CC * CC; i += 256) sW[i] = W[i];
    __syncthreads();

    int wid  = threadIdx.x >> 5;
    int lane = threadIdx.x & 31;
    int tile = blockIdx.x * 8 + wid;
    if (tile >= BB * TPB) return;
    int b    = tile / TPB;
    int row0 = (tile % TPB) * 16;

    int m  = lane & 15;   // A-row / B-col owned by this lane
    int kh = lane >> 4;   // which K-half (0 -> K 0,1 ; 1 -> K 2,3)

    v8f c0 = {}, c1 = {}, c2 = {}, c3 = {};
    const float* xb = x + ((size_t)b * NN + row0) * CC;

    for (int kk = 0; kk < CC; kk += 4) {
        int ka = kk + 2 * kh;
        v2f a;
        a.x = xb[m * CC + ka];
        a.y = xb[m * CC + ka + 1];
        v2f b0, b1, b2, b3;
        b0.x = sW[ka * CC +  0 + m];  b0.y = sW[(ka + 1) * CC +  0 + m];
        b1.x = sW[ka * CC + 16 + m];  b1.y = sW[(ka + 1) * CC + 16 + m];
        b2.x = sW[ka * CC + 32 + m];  b2.y = sW[(ka + 1) * CC + 32 + m];
        b3.x = sW[ka * CC + 48 + m];  b3.y = sW[(ka + 1) * CC + 48 + m];
        c0 = __builtin_amdgcn_wmma_f32_16x16x4_f32(false, a, false, b0, (short)0, c0, false, false);
        c1 = __builtin_amdgcn_wmma_f32_16x16x4_f32(false, a, false, b1, (short)0, c1, false, false);
        c2 = __builtin_amdgcn_wmma_f32_16x16x4_f32(false, a, false, b2, (short)0, c2, false, false);
        c3 = __builtin_amdgcn_wmma_f32_16x16x4_f32(false, a, false, b3, (short)0, c3, false, false);
    }

    float* hb = h + ((size_t)b * NN + row0) * CC;
    for (int r = 0; r < 8; ++r) {
        int row = r + 8 * kh;   // C layout: lanes 0-15 hold M=r, lanes 16-31 hold M=r+8
        hb[row * CC +  0 + m] = c0[r];
        hb[row * CC + 16 + m] = c1[r];
        hb[row * CC + 32 + m] = c2[r];
        hb[row * CC + 48 + m] = c3[r];
    }
}

// ---- per-node attention scalars a_s = h.att_src, a_d = h.att_dst ----
__global__ void k_asad(const float* __restrict__ h, const float* __restrict__ att_s,
                       const float* __restrict__ att_d,
                       float* __restrict__ a_s, float* __restrict__ a_d) {
    int idx = blockIdx.x * blockDim.x + threadIdx.x;
    if (idx >= BB * NN) return;
    const float* hr = h + (size_t)idx * CC;
    float s = 0.f, d = 0.f;
    for (int c = 0; c < CC; ++c) {
        float v = hr[c];
        s += v * att_s[c];
        d += v * att_d[c];
    }
    a_s[idx] = s;
    a_d[idx] = d;
}

__device__ __forceinline__ void edge_sd(const int* __restrict__ eidx, int i,
                                        int& src, int& dst) {
    if (i < EE) { src = eidx[i]; dst = eidx[EE + i]; }
    else        { src = i - EE; dst = i - EE; }
}

__device__ __forceinline__ float edge_logit(const float* a_s, const float* a_d,
                                            const float* a_edge, int b, int i,
                                            int src, int dst) {
    float lg = a_s[b * NN + src] + a_d[b * NN + dst] + a_edge[i];
    return (lg > 0.f) ? lg : 0.2f * lg;
}

// ---- pass 1: segment max of logits (ordered-uint atomicMax) ----
__global__ void k_max(const int* __restrict__ eidx, const float* __restrict__ a_s,
                      const float* __restrict__ a_d, const float* __restrict__ a_edge,
                      unsigned* __restrict__ muint) {
    long long gid = (long long)blockIdx.x * blockDim.x + threadIdx.x;
    if (gid >= (long long)BB * ET) return;
    int b = (int)(gid / ET), i = (int)(gid % ET);
    int src, dst;
    edge_sd(eidx, i, src, dst);
    float lg = edge_logit(a_s, a_d, a_edge, b, i, src, dst);
    atomicMax(&muint[b * NN + dst], f2o(lg));
}

// ---- pass 2: denom = segment sum of exp(logit - m) ----
__global__ void k_expsum(const int* __restrict__ eidx, const float* __restrict__ a_s,
                         const float* __restrict__ a_d, const float* __restrict__ a_edge,
                         const unsigned* __restrict__ muint, float* __restrict__ denom) {
    long long gid = (long long)blockIdx.x * blockDim.x + threadIdx.x;
    if (gid >= (long long)BB * ET) return;
    int b = (int)(gid / ET), i = (int)(gid % ET);
    int src, dst;
    edge_sd(eidx, i, src, dst);
    float lg = edge_logit(a_s, a_d, a_edge, b, i, src, dst);
    float ex = __expf(lg - o2f(muint[b * NN + dst]));
    atomicAdd(&denom[b * NN + dst], ex);
}

// ---- pass 3: out[dst] += alpha * h[src]  (one wave per edge, 2 chans/lane) ----
__global__ void __launch_bounds__(256) k_scatter(const int* __restrict__ eidx,
                                                 const float* __restrict__ a_s,
                                                 const float* __restrict__ a_d,
                                                 const float* __restrict__ a_edge,
                                                 const unsigned* __restrict__ muint,
                                                 const float* __restrict__ denom,
                                                 const float* __restrict__ h,
                                                 float* __restrict__ out) {
    long long gtid = (long long)blockIdx.x * blockDim.x + threadIdx.x;
    long long gw   = gtid >> 5;
    int lane = (int)(gtid & 31);
    if (gw >= (long long)BB * ET) return;
    int b = (int)(gw / ET), i = (int)(gw % ET);
    int src, dst;
    edge_sd(eidx, i, src, dst);
    float lg = edge_logit(a_s, a_d, a_edge, b, i, src, dst);
    int nd = b * NN + dst;
    float alpha = __expf(lg - o2f(muint[nd])) / denom[nd];
    const float* hs = h + ((size_t)b * NN + src) * CC;
    float* od = out + (size_t)nd * CC;
    int c = lane * 2;
    atomicAdd(&od[c],     hs[c]     * alpha);
    atomicAdd(&od[c + 1], hs[c + 1] * alpha);
}

// ---- epilogue: out = elu(out + bias) ----
__global__ void k_finish(float* __restrict__ out, const float* __restrict__ bias) {
    long long idx = (long long)blockIdx.x * blockDim.x + threadIdx.x;
    if (idx >= (long long)BB * NN * CC) return;
    float v = out[idx] + bias[(int)(idx & (CC - 1))];
    out[idx] = (v > 0.f) ? v : (__expf(v) - 1.f);
}

extern "C" void kernel_launch(void* const* d_in, const int* in_sizes, int n_in,
                              void* d_out, int out_size, void* d_ws, size_t ws_size,
                              hipStream_t stream) {
    const float* x        = (const float*)d_in[0];
    const int*   eidx     = (const int*)d_in[1];
    const float* ea       = (const float*)d_in[2];
    const float* W1       = (const float*)d_in[3];
    const float* W2       = (const float*)d_in[4];
    const float* W        = (const float*)d_in[5];
    const float* linW     = (const float*)d_in[6];
    const float* att_src  = (const float*)d_in[7];
    const float* att_dst  = (const float*)d_in[8];
    const float* att_edge = (const float*)d_in[9];
    const float* bias     = (const float*)d_in[10];
    float* out = (float*)d_out;

    float* ws = (float*)d_ws;
    size_t off = 0;
    float*    e_ws   = ws + off; off += (size_t)EE * EH;
    float*    lacc   = ws + off; off += (size_t)NN * EH;
    float*    cnt    = ws + off; off += NN;
    float*    a_edge = ws + off; off += ET;
    float*    v_edge = ws + off; off += 16;
    float*    h      = ws + off; off += (size_t)BB * NN * CC;
    float*    a_s    = ws + off; off += (size_t)BB * NN;
    float*    a_d    = ws + off; off += (size_t)BB * NN;
    unsigned* muint  = (unsigned*)(ws + off); off += (size_t)BB * NN;
    float*    denom  = ws + off; off += (size_t)BB * NN;

    hipMemsetAsync(cnt,   0, NN * sizeof(float), stream);
    hipMemsetAsync(lacc,  0, (size_t)NN * EH * sizeof(float), stream);
    hipMemsetAsync(muint, 0, (size_t)BB * NN * sizeof(unsigned), stream);  // 0 == -inf in ordered domain
    hipMemsetAsync(denom, 0, (size_t)BB * NN * sizeof(float), stream);
    hipMemsetAsync(out,   0, (size_t)BB * NN * CC * sizeof(float), stream);

    k_vedge<<<1, 64, 0, stream>>>(linW, att_edge, v_edge);
    k_edge_mlp<<<(EE + 255) / 256, 256, 0, stream>>>(ea, eidx, W1, W2, v_edge,
                                                     e_ws, cnt, lacc, a_edge);
    k_loop<<<(NN + 255) / 256, 256, 0, stream>>>(lacc, cnt, v_edge, a_edge);

    k_h_gemm<<<(BB * TPB + 7) / 8, 256, 0, stream>>>(x, W, h);
    k_asad<<<(BB * NN + 255) / 256, 256, 0, stream>>>(h, att_src, att_dst, a_s, a_d);

    long long nedge = (long long)BB * ET;
    k_max<<<(unsigned)((nedge + 255) / 256), 256, 0, stream>>>(eidx, a_s, a_d, a_edge, muint);
    k_expsum<<<(unsigned)((nedge + 255) / 256), 256, 0, stream>>>(eidx, a_s, a_d, a_edge, muint, denom);

    long long sth = nedge * 32;
    k_scatter<<<(unsigned)((sth + 255) / 256), 256, 0, stream>>>(eidx, a_s, a_d, a_edge,
                                                                 muint, denom, h, out);

    long long tot = (long long)BB * NN * CC;
    k_finish<<<(unsigned)((tot + 255) / 256), 256, 0, stream>>>(out, bias);
}